// WindowAttention_10075993277040
// MI455X (gfx1250) — compile-verified
//
#include <hip/hip_runtime.h>
#include <math.h>

typedef __attribute__((ext_vector_type(2))) float v2f;
typedef __attribute__((ext_vector_type(8))) float v8f;
typedef __attribute__((ext_vector_type(4))) float f4;

#define EPS 1e-5f

__device__ __forceinline__ v8f wmma4(v2f a, v2f b, v8f c) {
  // D(16x16,f32) = A(16x4,f32) * B(4x16,f32) + C
  return __builtin_amdgcn_wmma_f32_16x16x4_f32(false, a, false, b, (short)0, c,
                                               false, false);
}

// 16x64 (LDS, row-major) @ W[n,:].T  -> 16x16 C-layout tile, K=64 (16 chained WMMAs)
__device__ __forceinline__ v8f gemm16x64(const float* __restrict__ As,
                                         const float* __restrict__ W, int n0,
                                         float cinit, int lm, int lh) {
  v8f acc = {cinit, cinit, cinit, cinit, cinit, cinit, cinit, cinit};
  const float* arow = As + lm * 64 + 2 * lh;          // A: m=lm, k = kb + 2*lh + {0,1}
  const float* wrow = W + (n0 + lm) * 64 + 2 * lh;    // B[k][n] = W[n][k]
#pragma unroll
  for (int kb = 0; kb < 64; kb += 4) {
    v2f a = *(const v2f*)(arow + kb);
    v2f b = *(const v2f*)(wrow + kb);
    acc = wmma4(a, b, acc);
  }
  return acc;
}

// store C-layout tile: VGPR i holds row (i + 8*lh), column lm
__device__ __forceinline__ void store_tile(float* dst, int stride, int n0, v8f c,
                                           int lm, int lh) {
#pragma unroll
  for (int i = 0; i < 8; ++i) dst[(i + 8 * lh) * stride + n0 + lm] = c[i];
}

struct Params {
  const float *x, *p_r, *W1, *b1, *g1, *be1, *m1, *v1;
  const float *Wq, *bq, *Wk, *bk, *Wv, *bv;
  const float *Wp1, *bp1, *gp, *bep, *mp, *vp, *Wp2, *bp2;
  const float *Wd1, *bd1, *Wd2, *bd2;
  const float *go, *bo, *mo, *vo, *Wt, *bt, *ga, *ba, *ma, *va;
  float* out;
};

__global__ __launch_bounds__(256) void window_attn_kernel(Params p) {
  __shared__ float xs[16 * 64];    // input window (also the shortcut)
  __shared__ float h1s[16 * 64];   // h1; later reused as (shortcut - outbn)
  __shared__ float qs[16 * 64];
  __shared__ float ks_[16 * 64];
  __shared__ float vs[16 * 64];
  __shared__ float ps[256 * 16];   // positional bias p[k1][k2][d]
  __shared__ float att[256 * 16];  // per-head attention scratch
  __shared__ float s1b[8 * 256];   // per-wave GEMM1->GEMM2 transpose scratch
  __shared__ float outs[16 * 64];  // attention output (K x C)

  const int tid = threadIdx.x;
  const int wave = tid >> 5;
  const int lane = tid & 31;
  const int lm = lane & 15;  // M-row (A) / N-col (B,C)
  const int lh = lane >> 4;  // K half-select
  const int blk = blockIdx.x;  // b*N + n

  // ---- Phase A: stage x window into LDS (16x64 = 4KB) ----
  {
    const float* xg = p.x + (size_t)blk * 1024;
    *(f4*)(xs + tid * 4) = *(const f4*)(xg + tid * 4);
  }

  // ---- Phase B: positional MLP, thread t -> row (k1,k2)=t ----
  {
    const float* pr = p.p_r + (size_t)blk * 768 + tid * 3;
    float i0 = pr[0], i1 = pr[1], i2 = pr[2];
    float e0, e1, e2;
#pragma unroll
    for (int j = 0; j < 3; ++j) {
      float y = p.Wp1[j * 3 + 0] * i0 + p.Wp1[j * 3 + 1] * i1 +
                p.Wp1[j * 3 + 2] * i2 + p.bp1[j];
      float sc = p.gp[j] * rsqrtf(p.vp[j] + EPS);
      float e = fmaxf((y - p.mp[j]) * sc + p.bep[j], 0.f);
      if (j == 0) e0 = e; else if (j == 1) e1 = e; else e2 = e;
    }
#pragma unroll
    for (int d = 0; d < 16; ++d)
      ps[tid * 16 + d] = p.Wp2[d * 3 + 0] * e0 + p.Wp2[d * 3 + 1] * e1 +
                         p.Wp2[d * 3 + 2] * e2 + p.bp2[d];
  }
  __syncthreads();

  // ---- Phase C: h1 = relu(bn(x @ W1.T + b1)), 4 N-tiles on waves 0-3 ----
  if (wave < 4) {
    int n0 = wave * 16, n = n0 + lm;
    float sc = p.g1[n] * rsqrtf(p.v1[n] + EPS);
    float mm = p.m1[n], bb = p.be1[n];
    v8f c = gemm16x64(xs, p.W1, n0, p.b1[n], lm, lh);
#pragma unroll
    for (int i = 0; i < 8; ++i) c[i] = fmaxf((c[i] - mm) * sc + bb, 0.f);
    store_tile(h1s, 64, n0, c, lm, lh);
  }
  __syncthreads();

  // ---- Phase D: q,k,v = h1 @ W{q,k,v}.T + b (12 tiles over 8 waves) ----
  for (int tt = wave; tt < 12; tt += 8) {
    int which = tt >> 2;
    int n0 = (tt & 3) * 16, n = n0 + lm;
    const float* W = which == 0 ? p.Wq : (which == 1 ? p.Wk : p.Wv);
    const float* bias = which == 0 ? p.bq : (which == 1 ? p.bk : p.bv);
    float* dst = which == 0 ? qs : (which == 1 ? ks_ : vs);
    v8f c = gemm16x64(h1s, W, n0, bias[n], lm, lh);
    store_tile(dst, 64, n0, c, lm, lh);
  }

  // ---- Preload Wd1/Wd2 in B-layout VGPRs (reused for all 128 tiles) ----
  v2f w1b[4], w2b[4];
#pragma unroll
  for (int s = 0; s < 4; ++s) {
    w1b[s] = *(const v2f*)(p.Wd1 + lm * 16 + 4 * s + 2 * lh);
    w2b[s] = *(const v2f*)(p.Wd2 + lm * 16 + 4 * s + 2 * lh);
  }
  const float bd1n = p.bd1[lm], bd2n = p.bd2[lm];
  float* s1 = s1b + wave * 256;
  __syncthreads();

  // ---- Phase E: per-head vector attention ----
  for (int hd = 0; hd < 4; ++hd) {
    // each wave owns M-tiles k1 = wave, wave+8 (rows of a tile = k2)
    for (int k1 = wave; k1 < 16; k1 += 8) {
      const float* qrow = qs + k1 * 64 + hd * 16 + 2 * lh;   // depends on k1 only
      const float* krow = ks_ + lm * 64 + hd * 16 + 2 * lh;  // row m = k2 = lm
      const float* prow = ps + (k1 * 16 + lm) * 16 + 2 * lh;
      v8f a1 = {bd1n, bd1n, bd1n, bd1n, bd1n, bd1n, bd1n, bd1n};
#pragma unroll
      for (int s = 0; s < 4; ++s) {
        v2f qv = *(const v2f*)(qrow + 4 * s);
        v2f kv = *(const v2f*)(krow + 4 * s);
        v2f pv = *(const v2f*)(prow + 4 * s);
        v2f a = qv - kv + pv;  // u[k2][dk] = q[k1]-k[k2]+p[k1][k2]
        a1 = wmma4(a, w1b[s], a1);
      }
      // relu, lane-transpose via same-wave LDS scratch (DS ops are in-order)
#pragma unroll
      for (int i = 0; i < 8; ++i)
        s1[(i + 8 * lh) * 16 + lm] = fmaxf(a1[i], 0.f);
      const float* s1row = s1 + lm * 16 + 2 * lh;
      v8f a2 = {bd2n, bd2n, bd2n, bd2n, bd2n, bd2n, bd2n, bd2n};
#pragma unroll
      for (int s = 0; s < 4; ++s) {
        v2f a = *(const v2f*)(s1row + 4 * s);
        a2 = wmma4(a, w2b[s], a2);
      }
      store_tile(att + k1 * 256, 16, 0, a2, lm, lh);  // att[(k1,k2),d]
    }
    __syncthreads();

    // softmax over d: one thread per (k1,k2) row
    {
      float* row = att + tid * 16;
      float e[16];
      float mx = row[0];
#pragma unroll
      for (int d = 1; d < 16; ++d) mx = fmaxf(mx, row[d]);
      float sm = 0.f;
#pragma unroll
      for (int d = 0; d < 16; ++d) { e[d] = __expf(row[d] - mx); sm += e[d]; }
      float inv = 1.f / sm;
#pragma unroll
      for (int d = 0; d < 16; ++d) row[d] = e[d] * inv;
    }
    __syncthreads();

    // k2-normalize + weighted-sum with v: one thread per (k1,d)
    {
      int k1 = tid >> 4, d = tid & 15;
      const float* vcol = vs + hd * 16 + d;
      float s = 0.f, sv = 0.f;
#pragma unroll
      for (int k2 = 0; k2 < 16; ++k2) {
        float a = att[(k1 * 16 + k2) * 16 + d];
        s += a;
        sv += a * vcol[k2 * 64];
      }
      outs[k1 * 64 + hd * 16 + d] = sv / (1e-9f + s);
    }
    __syncthreads();  // att reused by next head
  }

  // ---- Phase F: out-BN, residual transform, final GEMM + residual ----
  {
    int base = tid * 4;
#pragma unroll
    for (int j = 0; j < 4; ++j) {
      int idx = base + j, c = idx & 63;
      float sc = p.go[c] * rsqrtf(p.vo[c] + EPS);
      float ob = fmaxf((outs[idx] - p.mo[c]) * sc + p.bo[c], 0.f);
      h1s[idx] = xs[idx] - ob;  // (shortcut - out)
    }
  }
  __syncthreads();
  if (wave < 4) {
    int n0 = wave * 16, n = n0 + lm;
    float sc = p.ga[n] * rsqrtf(p.va[n] + EPS);
    float mm = p.ma[n], bb = p.ba[n];
    v8f c = gemm16x64(h1s, p.Wt, n0, p.bt[n], lm, lh);
    float* og = p.out + (size_t)blk * 1024;
#pragma unroll
    for (int i = 0; i < 8; ++i) {
      int row = i + 8 * lh;
      float t = fmaxf((c[i] - mm) * sc + bb, 0.f);
      og[row * 64 + n] = xs[row * 64 + n] + t;  // shortcut + t
    }
  }
}

extern "C" void kernel_launch(void* const* d_in, const int* in_sizes, int n_in,
                              void* d_out, int out_size, void* d_ws,
                              size_t ws_size, hipStream_t stream) {
  (void)n_in; (void)d_ws; (void)ws_size; (void)out_size;
  Params p;
  p.x = (const float*)d_in[0];   p.p_r = (const float*)d_in[1];
  p.W1 = (const float*)d_in[2];  p.b1 = (const float*)d_in[3];
  p.g1 = (const float*)d_in[4];  p.be1 = (const float*)d_in[5];
  p.m1 = (const float*)d_in[6];  p.v1 = (const float*)d_in[7];
  p.Wq = (const float*)d_in[8];  p.bq = (const float*)d_in[9];
  p.Wk = (const float*)d_in[10]; p.bk = (const float*)d_in[11];
  p.Wv = (const float*)d_in[12]; p.bv = (const float*)d_in[13];
  p.Wp1 = (const float*)d_in[14]; p.bp1 = (const float*)d_in[15];
  p.gp = (const float*)d_in[16]; p.bep = (const float*)d_in[17];
  p.mp = (const float*)d_in[18]; p.vp = (const float*)d_in[19];
  p.Wp2 = (const float*)d_in[20]; p.bp2 = (const float*)d_in[21];
  p.Wd1 = (const float*)d_in[22]; p.bd1 = (const float*)d_in[23];
  p.Wd2 = (const float*)d_in[24]; p.bd2 = (const float*)d_in[25];
  p.go = (const float*)d_in[26]; p.bo = (const float*)d_in[27];
  p.mo = (const float*)d_in[28]; p.vo = (const float*)d_in[29];
  p.Wt = (const float*)d_in[30]; p.bt = (const float*)d_in[31];
  p.ga = (const float*)d_in[32]; p.ba = (const float*)d_in[33];
  p.ma = (const float*)d_in[34]; p.va = (const float*)d_in[35];
  p.out = (float*)d_out;

  int windows = in_sizes[0] / 1024;  // B*N (K*C = 16*64 = 1024)
  hipLaunchKernelGGL(window_attn_kernel, dim3(windows), dim3(256), 0, stream, p);
}